// RMILoss_14894946583030
// MI455X (gfx1250) — compile-verified
//
#include <hip/hip_runtime.h>
#include <hip/hip_bf16.h>

// ---------------------------------------------------------------------------
// RMI loss forward for MI455X (gfx1250, wave32).
// K2 uses V_WMMA_F32_16X16X4_F32 to accumulate the three 9x9 Gram matrices
// (labels x labels, probs x probs, labels x probs) over m = 169*169 samples,
// with a padded ones-row so the same WMMA tile also produces row sums for
// exact mean-centering. Phase-B hot loop is branch-free: 4 ds_load_b32 +
// 4 FMA lane-masks + 3 WMMAs per K=4 chunk, addresses strength-reduced.
// ---------------------------------------------------------------------------

#define NCLS 21
#define NBATCH 4
#define HH 512
#define WW 512
#define PP 171              // pooled H/W:  floor((512+2-3)/3)+1
#define QQ 169              // crop size:   171 - (3-1)
#define MTOT (QQ * QQ)      // 28561 samples per (n,c)
#define HALF_D 9
#define POS_ALPHA 5e-4f
#define CLIP_MIN 1e-6f
#define NB1 1024            // BCE partial blocks
#define SMEM_PAD 8          // absorb stray (masked) tail reads

typedef float v2f __attribute__((ext_vector_type(2)));
typedef float v8f __attribute__((ext_vector_type(8)));

// ---------------------------------------------------------------------------
// Kernel 1: masked BCE-with-logits partial sums (per-block, deterministic).
// ---------------------------------------------------------------------------
__global__ __launch_bounds__(256)
void rmi_bce_kernel(const float* __restrict__ logits,
                    const int* __restrict__ labels,
                    float* __restrict__ bceP,
                    float* __restrict__ cntP) {
    const int tid = threadIdx.x;
    const int gsz = gridDim.x * blockDim.x;
    int gtid = blockIdx.x * blockDim.x + tid;
    const int PIX = NBATCH * HH * WW;

    float acc = 0.f, cnt = 0.f;
    for (int p = gtid; p < PIX; p += gsz) {
        if (p + gsz < PIX) {
            __builtin_prefetch(&labels[p + gsz], 0, 0);   // global_prefetch_b8
        }
        int n = p / (HH * WW);
        int hw = p - n * (HH * WW);
        int l = labels[p];
        float mk = (l < NCLS) ? 1.f : 0.f;
        const float* lg = logits + (size_t)n * NCLS * HH * WW + hw;
        float a = 0.f;
        #pragma unroll
        for (int c = 0; c < NCLS; ++c) {
            float x = lg[(size_t)c * HH * WW];
            float tgt = (mk != 0.f && l == c) ? 1.f : 0.f;
            // max(x,0) - x*t + log1p(exp(-|x|))
            a += fmaxf(x, 0.f) - x * tgt + log1pf(__expf(-fabsf(x)));
        }
        acc += a * mk;
        cnt += mk;
    }

    __shared__ float sA[256];
    __shared__ float sC[256];
    sA[tid] = acc; sC[tid] = cnt;
    __syncthreads();
    for (int s = 128; s > 0; s >>= 1) {
        if (tid < s) { sA[tid] += sA[tid + s]; sC[tid] += sC[tid + s]; }
        __syncthreads();
    }
    if (tid == 0) { bceP[blockIdx.x] = sA[0]; cntP[blockIdx.x] = sC[0]; }
}

// ---------------------------------------------------------------------------
// Kernel 2: one block per (n,c).
//  Phase A: fused sigmoid + 3x3/stride3/pad1 maxpool of probs and one-hot
//           labels into LDS (two 171x171 f32 planes).
//  Phase B: WMMA f32 16x16x4 Gram accumulation over m, ones-row padded;
//           centered 9x9 covariances written to workspace.
// ---------------------------------------------------------------------------
__global__ __launch_bounds__(256)
void rmi_gram_kernel(const float* __restrict__ logits,
                     const int* __restrict__ labels,
                     float* __restrict__ cov_out /* [84][3][81] */) {
    extern __shared__ float smem[];
    float* laS = smem;            // PP*PP pooled one-hot labels
    float* prS = smem + PP * PP;  // PP*PP pooled probs (+ SMEM_PAD slack)

    const int b = blockIdx.x;
    const int n = b / NCLS;
    const int c = b % NCLS;
    const int tid = threadIdx.x;

    const float* lg = logits + (size_t)(n * NCLS + c) * (HH * WW);
    const int* lb = labels + (size_t)n * (HH * WW);

    // ---- Phase A: fused maxpool --------------------------------------------
    // max(mask*sigmoid(x)) over window == anyMasked ? sigmoid(max masked x) : 0
    for (int idx = tid; idx < PP * PP; idx += blockDim.x) {
        int oi = idx / PP, oj = idx - oi * PP;
        int r0 = 3 * oi - 1, c0 = 3 * oj - 1;
        float xmax = -1e30f;
        float anyv = 0.f;
        float lmax = 0.f;
        #pragma unroll
        for (int dy = 0; dy < 3; ++dy) {
            int r = r0 + dy;
            if (r < 0 || r >= HH) continue;
            #pragma unroll
            for (int dx = 0; dx < 3; ++dx) {
                int cc = c0 + dx;
                if (cc < 0 || cc >= WW) continue;
                int l = lb[r * WW + cc];
                float x = lg[r * WW + cc];
                if (l < NCLS) {
                    anyv = 1.f;
                    xmax = fmaxf(xmax, x);
                }
                if (l == c) lmax = 1.f;
            }
        }
        float s = 1.f / (1.f + __expf(-xmax));
        prS[idx] = anyv * s + CLIP_MIN;   // + CLIP_MIN commutes with max
        laS[idx] = lmax;
    }
    __syncthreads();

    // ---- Phase B: WMMA Gram accumulation (row-major, branch-free) ----------
    const int wave = tid >> 5;      // 8 waves
    const int lane = tid & 31;
    const int d = lane & 15;        // A row (M) == B col (N)
    const int khalf = lane >> 4;    // lanes 16-31 carry K=2,3
    const int sdy = (d < HALF_D) ? d / 3 : 0;
    const int sdx = (d < HALF_D) ? d % 3 : 0;
    const float rowsel = (d < HALF_D) ? 1.f : 0.f;   // real data rows 0..8
    const float onesel = (d == 15) ? 1.f : 0.f;      // ones row -> sums/count

    v8f accLL = {0.f,0.f,0.f,0.f,0.f,0.f,0.f,0.f};
    v8f accPP = {0.f,0.f,0.f,0.f,0.f,0.f,0.f,0.f};
    v8f accLP = {0.f,0.f,0.f,0.f,0.f,0.f,0.f,0.f};

    for (int i = wave; i < QQ; i += 8) {
        // lane-private base address for this sample row
        int a0 = (i + sdy) * PP + sdx + 2 * khalf;

        // 42 full chunks: columns j in [0,168), all valid -> no masks
        #pragma unroll 2
        for (int jc = 0; jc < 42; ++jc) {
            float l0 = laS[a0], l1 = laS[a0 + 1];
            float p0 = prS[a0], p1 = prS[a0 + 1];
            v2f la, pr;
            la.x = fmaf(l0, rowsel, onesel);
            la.y = fmaf(l1, rowsel, onesel);
            pr.x = fmaf(p0, rowsel, onesel);
            pr.y = fmaf(p1, rowsel, onesel);
            accLL = __builtin_amdgcn_wmma_f32_16x16x4_f32(false, la, false, la,
                                                          (short)0, accLL, false, false);
            accPP = __builtin_amdgcn_wmma_f32_16x16x4_f32(false, pr, false, pr,
                                                          (short)0, accPP, false, false);
            accLP = __builtin_amdgcn_wmma_f32_16x16x4_f32(false, la, false, pr,
                                                          (short)0, accLP, false, false);
            a0 += 4;
        }

        // tail chunk: j = 168 + 2*khalf + {0,1}; only j==168 is valid
        {
            int j0 = 168 + 2 * khalf;
            float v0 = (j0 < QQ) ? 1.f : 0.f;       // khalf==0 lanes only
            float l0 = laS[a0];                      // stray reads stay in padded LDS
            float p0 = prS[a0];
            v2f la, pr;
            la.x = v0 * fmaf(l0, rowsel, onesel);
            la.y = 0.f;
            pr.x = v0 * fmaf(p0, rowsel, onesel);
            pr.y = 0.f;
            accLL = __builtin_amdgcn_wmma_f32_16x16x4_f32(false, la, false, la,
                                                          (short)0, accLL, false, false);
            accPP = __builtin_amdgcn_wmma_f32_16x16x4_f32(false, pr, false, pr,
                                                          (short)0, accPP, false, false);
            accLP = __builtin_amdgcn_wmma_f32_16x16x4_f32(false, la, false, pr,
                                                          (short)0, accLP, false, false);
        }
    }

    // ---- cross-wave reduction (fixed order, reuse LDS) ---------------------
    float* red = smem;              // 8*256 floats, aliases pooled planes
    float g0, g1, g2;

    __syncthreads();
    #pragma unroll
    for (int r = 0; r < 8; ++r)
        red[wave * 256 + (r + 8 * khalf) * 16 + d] = accLL[r];
    __syncthreads();
    g0 = 0.f;
    for (int wv = 0; wv < 8; ++wv) g0 += red[wv * 256 + tid];
    __syncthreads();

    #pragma unroll
    for (int r = 0; r < 8; ++r)
        red[wave * 256 + (r + 8 * khalf) * 16 + d] = accPP[r];
    __syncthreads();
    g1 = 0.f;
    for (int wv = 0; wv < 8; ++wv) g1 += red[wv * 256 + tid];
    __syncthreads();

    #pragma unroll
    for (int r = 0; r < 8; ++r)
        red[wave * 256 + (r + 8 * khalf) * 16 + d] = accLP[r];
    __syncthreads();
    g2 = 0.f;
    for (int wv = 0; wv < 8; ++wv) g2 += red[wv * 256 + tid];
    __syncthreads();

    // ---- center and emit 9x9 covariances -----------------------------------
    float* G = smem;                // 3*256 floats
    G[0 * 256 + tid] = g0;
    G[1 * 256 + tid] = g1;
    G[2 * 256 + tid] = g2;
    __syncthreads();

    const int rd = tid / 16, re = tid % 16;
    if (rd < HALF_D && re < HALF_D) {
        const float Mf = (float)MTOT;
        float sl_d = G[0 * 256 + rd * 16 + 15];   // S_ll[d][15] = sum la_d
        float sl_e = G[0 * 256 + re * 16 + 15];
        float sp_d = G[1 * 256 + rd * 16 + 15];   // S_pp[d][15] = sum pr_d
        float sp_e = G[1 * 256 + re * 16 + 15];
        float covLL = G[0 * 256 + tid] - sl_d * sl_e / Mf;
        float covPP = G[1 * 256 + tid] - sp_d * sp_e / Mf;
        float covLP = G[2 * 256 + tid] - sl_d * sp_e / Mf;
        float* o = cov_out + (size_t)b * 243;
        o[0 * 81 + rd * 9 + re] = covLL;
        o[1 * 81 + rd * 9 + re] = covPP;
        o[2 * 81 + rd * 9 + re] = covLP;
    }
}

// ---------------------------------------------------------------------------
// Kernel 3: per-(n,c) 9x9 Cholesky/solve/logdet + deterministic final combine.
// ---------------------------------------------------------------------------
__device__ __forceinline__ void chol9(float* A) {
    #pragma unroll
    for (int k = 0; k < 9; ++k) {
        float dv = sqrtf(fmaxf(A[k * 9 + k], 1e-30f));
        A[k * 9 + k] = dv;
        float inv = 1.f / dv;
        for (int i = k + 1; i < 9; ++i) A[i * 9 + k] *= inv;
        for (int j = k + 1; j < 9; ++j)
            for (int i = j; i < 9; ++i)
                A[i * 9 + j] -= A[i * 9 + k] * A[j * 9 + k];
    }
}

__global__ __launch_bounds__(128)
void rmi_finalize(const float* __restrict__ cov,
                  const float* __restrict__ bceP,
                  const float* __restrict__ cntP,
                  float* __restrict__ out) {
    __shared__ float part[128];
    const int t = threadIdx.x;
    float contrib = 0.f;

    if (t < NBATCH * NCLS) {
        float la[81], pp[81], lp[81], Z[81], ap[81];
        const float* o = cov + (size_t)t * 243;
        for (int i = 0; i < 81; ++i) { la[i] = o[i]; pp[i] = o[81 + i]; lp[i] = o[162 + i]; }

        // pr_cov + alpha*I -> L L^T
        for (int k = 0; k < 9; ++k) pp[k * 9 + k] += POS_ALPHA;
        chol9(pp);

        // Z[:,a] = L^{-1} * la_pr_cov[a,:]^T  (forward substitution)
        for (int a = 0; a < 9; ++a) {
            for (int i = 0; i < 9; ++i) {
                float v = lp[a * 9 + i];
                for (int k = 0; k < i; ++k) v -= pp[i * 9 + k] * Z[a * 9 + k];
                Z[a * 9 + i] = v / pp[i * 9 + i];
            }
        }

        // appro_var = la_cov - Z^T Z, then + alpha*I and Cholesky logdet
        for (int a = 0; a < 9; ++a)
            for (int b2 = 0; b2 < 9; ++b2) {
                float s = 0.f;
                for (int i = 0; i < 9; ++i) s += Z[a * 9 + i] * Z[b2 * 9 + i];
                ap[a * 9 + b2] = la[a * 9 + b2] - s;
            }
        for (int k = 0; k < 9; ++k) ap[k * 9 + k] += POS_ALPHA;
        chol9(ap);

        float logdet = 0.f;
        for (int k = 0; k < 9; ++k) logdet += __logf(fmaxf(ap[k * 9 + k], 1e-30f));
        logdet *= 2.f;
        // rmi_now = 0.5*logdet;  sum over (n,c) of rmi_now / (NBATCH*HALF_D)
        contrib = 0.5f * logdet / ((float)NBATCH * (float)HALF_D);
    }
    part[t] = contrib;
    __syncthreads();

    if (t == 0) {
        float rmi = 0.f;
        for (int i = 0; i < NBATCH * NCLS; ++i) rmi += part[i];
        float sb = 0.f, sc = 0.f;
        for (int i = 0; i < NB1; ++i) { sb += bceP[i]; sc += cntP[i]; }
        float bce = sb / (sc + 1.f);
        // lambda_way=1, lambda=0.5: 0.5*bce + 0.5*rmi
        out[0] = 0.5f * bce + 0.5f * rmi;
    }
}

// ---------------------------------------------------------------------------
extern "C" void kernel_launch(void* const* d_in, const int* in_sizes, int n_in,
                              void* d_out, int out_size, void* d_ws, size_t ws_size,
                              hipStream_t stream) {
    const float* logits = (const float*)d_in[0];
    const int* labels = (const int*)d_in[1];
    float* out = (float*)d_out;
    float* ws = (float*)d_ws;

    float* bceP = ws;                 // NB1 floats
    float* cntP = ws + NB1;           // NB1 floats
    float* cov  = ws + 2 * NB1;       // 84 * 243 floats

    rmi_bce_kernel<<<NB1, 256, 0, stream>>>(logits, labels, bceP, cntP);

    size_t shmem = (size_t)(2 * PP * PP + SMEM_PAD) * sizeof(float); // ~228 KB of 320 KB WGP LDS
    rmi_gram_kernel<<<NBATCH * NCLS, 256, shmem, stream>>>(logits, labels, cov);

    rmi_finalize<<<1, 128, 0, stream>>>(cov, bceP, cntP, out);
}